// EvalWrapper_7937099563192
// MI455X (gfx1250) — compile-verified
//
#include <hip/hip_runtime.h>
#include <hip/hip_bf16.h>

#define B_      2
#define H_      64
#define W_      64
#define N_      (H_ * W_)      // 4096 pixels
#define VN_     9
#define HYP_    512
#define EPS_    1e-6f
#define THRESH_ 0.99f

typedef __attribute__((ext_vector_type(2))) float v2f;
typedef __attribute__((ext_vector_type(8))) float v8f;

// ---------------------------------------------------------------- RNG ------
__device__ __forceinline__ unsigned long long sm64(unsigned long long z) {
    z += 0x9E3779B97F4A7C15ULL;
    z = (z ^ (z >> 30)) * 0xBF58476D1CE4E5B9ULL;
    z = (z ^ (z >> 27)) * 0x94D049BB133111EBULL;
    return z ^ (z >> 31);
}

// ------------------------------------------------------------- prep --------
// Per pixel: mask = argmax(seg)==1; per keypoint, B-operand for the cosine
// numerator WMMA: [dnx, dny, -(p.dn), 0] with the mask folded in (-1e30 for
// background -> never an inlier).  Also the distance B-operand per pixel:
// [-2x, -2y, 1, x^2+y^2]  (so A=[hx,hy,|h|^2,1] gives |h-p|^2).
__global__ __launch_bounds__(256) void prep_kernel(
    const float* __restrict__ seg, const float* __restrict__ vp,
    float4* __restrict__ bnum, float4* __restrict__ b2) {
  int tid = blockIdx.x * blockDim.x + threadIdx.x;
  if (tid >= B_ * N_) return;
  int b = tid / N_, n = tid % N_;
  float x = (float)(n % W_), y = (float)(n / W_);
  const float* s = seg + (size_t)b * 2 * N_;
  bool m = s[N_ + n] > s[n];                       // argmax over 2 channels
  const float* vb = vp + (size_t)b * 2 * VN_ * N_;
#pragma unroll
  for (int v = 0; v < VN_; ++v) {
    float dx = vb[(2 * v) * N_ + n];
    float dy = vb[(2 * v + 1) * N_ + n];
    float inv = rsqrtf(dx * dx + dy * dy + EPS_);
    float dnx = dx * inv, dny = dy * inv;
    float4 o = m ? make_float4(dnx, dny, -(x * dnx + y * dny), 0.f)
                 : make_float4(0.f, 0.f, -1e30f, 0.f);
    bnum[(size_t)(b * VN_ + v) * N_ + n] = o;
  }
  if (b == 0) b2[n] = make_float4(-2.f * x, -2.f * y, 1.f, x * x + y * y);
}

// ------------------------------------------- foreground compaction ---------
// Deterministic (scan-based, no atomics): one block per batch image.
__global__ __launch_bounds__(256) void fg_kernel(
    const float* __restrict__ seg, int* __restrict__ fg, int* __restrict__ fgcnt) {
  int b = blockIdx.x, t = threadIdx.x;
  __shared__ int sc[256];
  const float* s0 = seg + (size_t)b * 2 * N_;
  const float* s1 = s0 + N_;
  int base = t * 16, cnt = 0;
#pragma unroll
  for (int i = 0; i < 16; ++i) cnt += (s1[base + i] > s0[base + i]) ? 1 : 0;
  sc[t] = cnt;
  __syncthreads();
  for (int off = 1; off < 256; off <<= 1) {        // Hillis-Steele inclusive scan
    int val = (t >= off) ? sc[t - off] : 0;
    __syncthreads();
    sc[t] += val;
    __syncthreads();
  }
  int w = sc[t] - cnt;                             // exclusive prefix
  int* out = fg + b * N_;
  for (int i = 0; i < 16; ++i) {
    int n = base + i;
    if (s1[n] > s0[n]) out[w++] = n;
  }
  if (t == 255) fgcnt[b] = sc[255];
  if (t == 0 && sc[255] == 0) out[0] = 0;          // safe fallback, deterministic
}

// -------------------------------------------------- hypothesis gen ---------
__global__ __launch_bounds__(256) void hyp_kernel(
    const float* __restrict__ vp, const int* __restrict__ fg,
    const int* __restrict__ fgcnt, float4* __restrict__ hyp) {
  int tid = blockIdx.x * blockDim.x + threadIdx.x;
  if (tid >= B_ * HYP_ * VN_) return;
  int b = tid / (HYP_ * VN_);
  int r = tid % (HYP_ * VN_);
  int v = r % VN_;
  unsigned long long cnt = (unsigned long long)max(fgcnt[b], 1);
  int i0 = fg[b * N_ + (int)(sm64(0x2A5Full ^ (unsigned long long)(2 * tid)) % cnt)] & (N_ - 1);
  int i1 = fg[b * N_ + (int)(sm64(0x2A5Full ^ (unsigned long long)(2 * tid + 1)) % cnt)] & (N_ - 1);
  float x0 = (float)(i0 % W_), y0 = (float)(i0 / W_);
  float x1 = (float)(i1 % W_), y1 = (float)(i1 / W_);
  const float* vb = vp + ((size_t)b * 2 * VN_ + 2 * v) * N_;
  float d0x = vb[i0], d0y = vb[N_ + i0];
  float d1x = vb[i1], d1y = vb[N_ + i1];
  float n0x = d0y, n0y = -d0x, n1x = d1y, n1y = -d1x;   // perp()
  float det = n0x * n1y - n0y * n1x;
  float dets = (fabsf(det) < EPS_) ? EPS_ : det;
  float c0 = n0x * x0 + n0y * y0;
  float c1 = n1x * x1 + n1y * y1;
  float hx = (c0 * n1y - c1 * n0y) / dets;
  float hy = (n0x * c1 - n1x * c0) / dets;
  hyp[(b * HYP_ + (r / VN_)) * VN_ + v] = make_float4(hx, hy, hx * hx + hy * hy, 1.f);
}

// --------------------------------------------------- WMMA voting -----------
// One wave32 per (b, v, 16-hypothesis tile).  Per 16-pixel tile: 2x
// v_wmma_f32_16x16x4_f32 produce the cosine numerator and squared distance
// for all 256 (hyp,pixel) pairs.
// inlier <=> num > T*sqrt(d2+eps) <=> (num > 0) && (num^2 > T^2*d2 + T^2*eps)
// -- squared form: no sqrt; single fused fma for the threshold (no fmax
//    canonicalization ops in the hot loop).
// A layout (16x4 f32): lane L holds row M=L%16, K pair 2*(L/16).
// B layout (4x16 f32): lane L holds col N=L%16, K pair 2*(L/16).
// C/D layout: lane L col N=L%16; VGPR r -> row r + 8*(L/16).
__global__ __launch_bounds__(256) void vote_kernel(
    const float4* __restrict__ bnum, const float4* __restrict__ b2,
    const float4* __restrict__ hyp, int* __restrict__ counts) {
  int wid  = (blockIdx.x * blockDim.x + threadIdx.x) >> 5;
  int lane = threadIdx.x & 31;
  int b    = wid / (VN_ * 32);
  int rem  = wid % (VN_ * 32);
  int v    = rem / 32;
  int tile = rem % 32;
  int half = lane >> 4;
  int lo   = lane & 15;

  float4 hp = hyp[(b * HYP_ + tile * 16 + lo) * VN_ + v];
  v2f aN, aD;                                   // A operands (no EXEC change)
  aN.x = half ? 1.f : hp.x;  aN.y = half ? 0.f : hp.y;
  aD.x = half ? hp.z : hp.x; aD.y = half ? 1.f : hp.y;

  const float T2    = THRESH_ * THRESH_;
  const float T2EPS = THRESH_ * THRESH_ * EPS_;

  int acc[8];
#pragma unroll
  for (int r = 0; r < 8; ++r) acc[r] = 0;

  const float4* bn_base = bnum + (size_t)(b * VN_ + v) * N_;
#pragma unroll 2
  for (int t = 0; t < N_ / 16; ++t) {
    int n = t * 16 + lo;
    float4 bn = bn_base[n];
    float4 q  = b2[n];
    if (t + 16 < N_ / 16) __builtin_prefetch(bn_base + n + 256, 0, 1);
    v2f bN, bD;
    bN.x = half ? bn.z : bn.x; bN.y = half ? 0.f : bn.y;
    bD.x = half ? q.z  : q.x;  bD.y = half ? q.w : q.y;
    v8f cz = {};
    v8f num = __builtin_amdgcn_wmma_f32_16x16x4_f32(false, aN, false, bN,
                                                    (short)0, cz, false, false);
    v8f d2  = __builtin_amdgcn_wmma_f32_16x16x4_f32(false, aD, false, bD,
                                                    (short)0, cz, false, false);
#pragma unroll
    for (int r = 0; r < 8; ++r) {
      float nm = num[r];
      float th = __builtin_fmaf(T2, d2[r], T2EPS);   // T^2*(d2+eps), one fma
      acc[r] += (nm > 0.f && nm * nm > th) ? 1 : 0;
    }
  }
  // reduce across the 16 lanes that share each hypothesis row
#pragma unroll
  for (int r = 0; r < 8; ++r)
    for (int off = 1; off < 16; off <<= 1)
      acc[r] += __shfl_xor(acc[r], off, 16);
  if (lo == 0) {
    int base = (b * VN_ + v) * HYP_ + tile * 16 + half * 8;
#pragma unroll
    for (int r = 0; r < 8; ++r) counts[base + r] = acc[r];
  }
}

// ----------------------------------------------------- argmax --------------
__global__ __launch_bounds__(256) void argmax_kernel(
    const int* __restrict__ counts, const float4* __restrict__ hyp,
    float4* __restrict__ winner) {
  int bv = blockIdx.x, t = threadIdx.x;
  int b = bv / VN_, v = bv % VN_;
  __shared__ int bc[256], bh[256];
  const int* c = counts + bv * HYP_;
  int c0 = c[t], c1 = c[t + 256];
  int bestc = (c1 > c0) ? c1 : c0;               // tie -> lower index (jnp.argmax)
  int besth = (c1 > c0) ? (t + 256) : t;
  bc[t] = bestc; bh[t] = besth;
  __syncthreads();
  for (int off = 128; off > 0; off >>= 1) {
    if (t < off) {
      if (bc[t + off] > bc[t] || (bc[t + off] == bc[t] && bh[t + off] < bh[t])) {
        bc[t] = bc[t + off]; bh[t] = bh[t + off];
      }
    }
    __syncthreads();
  }
  if (t == 0) winner[bv] = hyp[(b * HYP_ + bh[0]) * VN_ + v];
}

// ----------------------------------------------------- refine --------------
__global__ __launch_bounds__(256) void refine_kernel(
    const float* __restrict__ vp, const float4* __restrict__ bnum,
    const float4* __restrict__ winner, float* __restrict__ out) {
  int bv = blockIdx.x, t = threadIdx.x;
  int b = bv / VN_, v = bv % VN_;
  float4 wn = winner[bv];
  float wx = wn.x, wy = wn.y;
  const float4* bn  = bnum + (size_t)bv * N_;
  const float*  vbx = vp + ((size_t)b * 2 * VN_ + 2 * v) * N_;
  const float*  vby = vbx + N_;
  float a00 = 0, a01 = 0, a11 = 0, bb0 = 0, bb1 = 0;
  const float T2 = THRESH_ * THRESH_;
  for (int i = 0; i < 16; ++i) {
    int n = i * 256 + t;
    float x = (float)(n % W_), y = (float)(n / W_);
    float4 d = bn[n];                               // masked pixels: d.xy==0 -> cw==0
    float dwx = wx - x, dwy = wy - y;
    float d2 = dwx * dwx + dwy * dwy + EPS_;
    float cn = dwx * d.x + dwy * d.y;               // cw = cn / sqrt(d2)
    if (cn > 0.f && cn * cn > T2 * d2) {            // cw > THRESH, sqrt-free
      float dx = vbx[n], dy = vby[n];
      float nx = dy, ny = -dx;                      // perp(raw direct)
      float bb = nx * x + ny * y;
      a00 += nx * nx; a01 += nx * ny; a11 += ny * ny;
      bb0 += nx * bb; bb1 += ny * bb;
    }
  }
  __shared__ float red[5][256];
  red[0][t] = a00; red[1][t] = a01; red[2][t] = a11; red[3][t] = bb0; red[4][t] = bb1;
  __syncthreads();
  for (int off = 128; off > 0; off >>= 1) {
    if (t < off)
#pragma unroll
      for (int k = 0; k < 5; ++k) red[k][t] += red[k][t + off];
    __syncthreads();
  }
  if (t == 0) {
    a00 = red[0][0]; a01 = red[1][0]; a11 = red[2][0]; bb0 = red[3][0]; bb1 = red[4][0];
    float detA = a00 * a11 - a01 * a01;
    float dets = (fabsf(detA) < EPS_) ? EPS_ : detA;
    float px = ( a11 * bb0 - a01 * bb1) / dets;
    float py = (-a01 * bb0 + a00 * bb1) / dets;
    bool ok = fabsf(detA) >= EPS_;
    out[bv * 2 + 0] = ok ? px : wx;
    out[bv * 2 + 1] = ok ? py : wy;
  }
}

// ------------------------------------------------------------ launch -------
extern "C" void kernel_launch(void* const* d_in, const int* in_sizes, int n_in,
                              void* d_out, int out_size, void* d_ws, size_t ws_size,
                              hipStream_t stream) {
  const float* seg = (const float*)d_in[0];   // [B,2,H,W]
  const float* vp  = (const float*)d_in[1];   // [B,2*VN,H,W]
  float* out = (float*)d_out;                 // [B,VN,2]

  char* ws = (char*)d_ws;                     // deterministic layout, ~1.5 MB
  int*    fgcnt  = (int*)ws;                                    // 64 B slot
  int*    fg     = (int*)(ws + 64);                             // B*N ints
  float4* bnum   = (float4*)(ws + 64 + (size_t)B_ * N_ * 4);    // B*VN*N
  float4* b2     = bnum + (size_t)B_ * VN_ * N_;                // N
  float4* hyp    = b2 + N_;                                     // B*HYP*VN
  int*    counts = (int*)(hyp + (size_t)B_ * HYP_ * VN_);       // B*VN*HYP
  float4* winner = (float4*)(counts + B_ * VN_ * HYP_);         // B*VN

  prep_kernel<<<(B_ * N_ + 255) / 256, 256, 0, stream>>>(seg, vp, bnum, b2);
  fg_kernel<<<B_, 256, 0, stream>>>(seg, fg, fgcnt);
  hyp_kernel<<<(B_ * HYP_ * VN_ + 255) / 256, 256, 0, stream>>>(vp, fg, fgcnt, hyp);
  vote_kernel<<<(B_ * VN_ * 32) / 8, 256, 0, stream>>>(bnum, b2, hyp, counts);
  argmax_kernel<<<B_ * VN_, 256, 0, stream>>>(counts, hyp, winner);
  refine_kernel<<<B_ * VN_, 256, 0, stream>>>(vp, bnum, winner, out);
}